// MaxPoolingMatching_6717328851298
// MI455X (gfx1250) — compile-verified
//
#include <hip/hip_runtime.h>

typedef __attribute__((ext_vector_type(2))) float v2f;
typedef __attribute__((ext_vector_type(4))) float v4f;
typedef __attribute__((ext_vector_type(8))) float v8f;

#define BB 16
#define LL 256
#define DD 256
#define PP 20
#define SLAB_STRIDE 20   // 16 data floats + 4 pad: conflict-free (gcd(20,64) spread) & 16B aligned

__device__ __forceinline__ v8f wmma4(v2f a, v2f b, v8f c) {
  // D = A(16x4, f32) * B(4x16, f32) + C(16x16, f32)
  return __builtin_amdgcn_wmma_f32_16x16x4_f32(false, a, false, b, (short)0, c, false, false);
}

__global__ __launch_bounds__(256)
void mpm_kernel(const float* __restrict__ s1, const float* __restrict__ s2,
                const float* __restrict__ kern, float* __restrict__ out) {
  __shared__ float wsq[DD];                 // kernel[p,:]^2
  __shared__ float inv1[LL];                // 1/||s1[b,l]*k_p||
  __shared__ float inv2[LL];                // 1/||s2[b,m]*k_p||
  __shared__ float slab[LL * SLAB_STRIDE];  // (s2[b,m,d]*wsq[d]) tile: [m][k-within-slab]

  const int b    = blockIdx.x / PP;
  const int p    = blockIdx.x % PP;
  const int tid  = threadIdx.x;
  const int lane = tid & 31;
  const int wave = tid >> 5;
  const int hi   = lane >> 4;   // 0 or 1 (lane half)
  const int ln   = lane & 15;

  // ---- stage kernel row squared ----
  {
    float kv = kern[p * DD + tid];
    wsq[tid] = kv * kv;
  }
  __syncthreads();

  // ---- inverse norms: wave w handles rows [w*32, w*32+32) ----
  {
    const v4f* wq = (const v4f*)wsq;
    for (int i = 0; i < 32; ++i) {
      int row = wave * 32 + i;
      const v4f* x1 = (const v4f*)(s1 + ((size_t)b * LL + row) * DD);
      const v4f* x2 = (const v4f*)(s2 + ((size_t)b * LL + row) * DD);
      float a1 = 0.f, a2 = 0.f;
      #pragma unroll
      for (int c = 0; c < 2; ++c) {   // lane covers d = lane*8 .. lane*8+7
        v4f xa = x1[lane * 2 + c];
        v4f xb = x2[lane * 2 + c];
        v4f w4 = wq[lane * 2 + c];
        v4f t1 = xa * xa * w4;
        v4f t2 = xb * xb * w4;
        a1 += t1.x + t1.y + t1.z + t1.w;
        a2 += t2.x + t2.y + t2.z + t2.w;
      }
      #pragma unroll
      for (int off = 16; off > 0; off >>= 1) {
        a1 += __shfl_xor(a1, off);
        a2 += __shfl_xor(a2, off);
      }
      if (lane == 0) {
        inv1[row] = rsqrtf(fmaxf(a1, 1e-12f));
        inv2[row] = rsqrtf(fmaxf(a2, 1e-12f));
      }
    }
  }
  __syncthreads();

  // ---- main: per wave, 2 row-tiles (32 rows) x 8 m-tiles per m-group ----
  const int wrow0 = wave * 32;
  float rmax[2][8];
  #pragma unroll
  for (int rt = 0; rt < 2; ++rt)
    #pragma unroll
    for (int v = 0; v < 8; ++v) rmax[rt][v] = -3.402823466e38f;

  for (int mg = 0; mg < 2; ++mg) {
    v8f acc[2][8];
    #pragma unroll
    for (int rt = 0; rt < 2; ++rt)
      #pragma unroll
      for (int mt = 0; mt < 8; ++mt) acc[rt][mt] = (v8f){0.f,0.f,0.f,0.f,0.f,0.f,0.f,0.f};

    for (int kslab = 0; kslab < 16; ++kslab) {
      const int k0 = kslab * 16;
      __syncthreads();   // previous slab fully consumed
      // cooperative fill: thread t stages m-row t (16 k-values), wsq folded in
      {
        const v4f* src = (const v4f*)(s2 + ((size_t)b * LL + tid) * DD + k0);
        const v4f* wq  = (const v4f*)(wsq + k0);
        float* dst = slab + tid * SLAB_STRIDE;
        #pragma unroll
        for (int c = 0; c < 4; ++c) {
          v4f g = src[c] * wq[c];
          *(v4f*)(dst + 4 * c) = g;
        }
      }
      __syncthreads();

      #pragma unroll
      for (int ks = 0; ks < 4; ++ks) {
        const int k = k0 + ks * 4;
        // A fragments: raw sent1, M = ln, K = k + 2*hi + {0,1}
        v2f afrag[2];
        #pragma unroll
        for (int rt = 0; rt < 2; ++rt) {
          const float* ap = s1 + ((size_t)b * LL + wrow0 + rt * 16 + ln) * DD + k + 2 * hi;
          afrag[rt] = *(const v2f*)ap;
        }
        #pragma unroll
        for (int mt = 0; mt < 8; ++mt) {
          const int m = mg * 128 + mt * 16 + ln;   // N = ln
          v2f bfrag = *(const v2f*)(slab + m * SLAB_STRIDE + ks * 4 + 2 * hi);
          acc[0][mt] = wmma4(afrag[0], bfrag, acc[0][mt]);
          acc[1][mt] = wmma4(afrag[1], bfrag, acc[1][mt]);
        }
      }
    }

    // fold this m-group into the running max (scale columns by inv_n2)
    #pragma unroll
    for (int mt = 0; mt < 8; ++mt) {
      float sc = inv2[mg * 128 + mt * 16 + ln];
      #pragma unroll
      for (int rt = 0; rt < 2; ++rt)
        #pragma unroll
        for (int v = 0; v < 8; ++v)
          rmax[rt][v] = fmaxf(rmax[rt][v], acc[rt][mt][v] * sc);
    }
  }

  // ---- reduce max over the 16 lanes holding one row's N values ----
  #pragma unroll
  for (int rt = 0; rt < 2; ++rt)
    #pragma unroll
    for (int v = 0; v < 8; ++v) {
      float x = rmax[rt][v];
      #pragma unroll
      for (int off = 8; off > 0; off >>= 1)
        x = fmaxf(x, __shfl_xor(x, off));
      rmax[rt][v] = x;
    }

  if (ln == 0) {  // lanes 0 and 16 each write 16 rows (hi selects M upper half)
    #pragma unroll
    for (int rt = 0; rt < 2; ++rt)
      #pragma unroll
      for (int v = 0; v < 8; ++v) {
        int row = wrow0 + rt * 16 + v + 8 * hi;
        out[((size_t)b * LL + row) * PP + p] = rmax[rt][v] * inv1[row];
      }
  }
}

extern "C" void kernel_launch(void* const* d_in, const int* in_sizes, int n_in,
                              void* d_out, int out_size, void* d_ws, size_t ws_size,
                              hipStream_t stream) {
  const float* s1   = (const float*)d_in[0];  // [16,256,256] f32
  const float* s2   = (const float*)d_in[1];  // [16,256,256] f32
  const float* kern = (const float*)d_in[2];  // [20,256]     f32
  float* out = (float*)d_out;                 // [16,256,20]  f32
  (void)in_sizes; (void)n_in; (void)out_size; (void)d_ws; (void)ws_size;
  dim3 grid(BB * PP), block(256);
  hipLaunchKernelGGL(mpm_kernel, grid, block, 0, stream, s1, s2, kern, out);
}